// SupConMultiClassLoss_25220047962607
// MI455X (gfx1250) — compile-verified
//
#include <hip/hip_runtime.h>

typedef float fx2 __attribute__((ext_vector_type(2)));
typedef float fx8 __attribute__((ext_vector_type(8)));

#define DIM      128
#define K_STEPS  32          // DIM / 4 (K=4 per f32 WMMA)
#define PC       64          // columns staged per LDS panel
#define PSTRIDE  132         // padded floats per column (conflict-free ds_load_b64)
#define WAVES    2
#define BLOCK    (WAVES * 32)
#define ROWS_PER_WG (WAVES * 16)
#define INV_TAU  10.0f
#define SENT     (-1e30f)
#define NC       100         // number of classes (reference: C=100)

// ---------------- Pre-pass: per-class feature sums + counts ----------------
__global__ __launch_bounds__(DIM)
void class_sums(const float* __restrict__ z, const int* __restrict__ labels,
                float* __restrict__ csum, float* __restrict__ ccnt, int rowsPerWG)
{
    __shared__ float tbl[NC * DIM];
    __shared__ float cnt[NC];
    const int t = threadIdx.x;
    for (int c = 0; c < NC; ++c) tbl[c * DIM + t] = 0.f;
    if (t < NC) cnt[t] = 0.f;
    __syncthreads();

    const int r0 = blockIdx.x * rowsPerWG;
    for (int r = r0; r < r0 + rowsPerWG; ++r) {
        int lab = labels[r];
        tbl[lab * DIM + t] += z[(size_t)r * DIM + t];
        if (t == 0) cnt[lab] += 1.f;
    }
    __syncthreads();

    for (int c = 0; c < NC; ++c) atomicAdd(&csum[c * DIM + t], tbl[c * DIM + t]);
    if (t < NC) atomicAdd(&ccnt[t], cnt[t]);
}

// ---------------- Main fused sim + online-LSE kernel ----------------
__global__ __launch_bounds__(BLOCK)
void supcon_main(const float* __restrict__ z, const int* __restrict__ labels,
                 const float* __restrict__ csum, const float* __restrict__ ccnt,
                 float* __restrict__ acc, int B)
{
    __shared__ float panel[PC * PSTRIDE];

    const int tid  = threadIdx.x;
    const int lane = tid & 31;           // wave32
    const int wv   = tid >> 5;
    const int n    = lane & 15;          // column residue within tile
    const int half = lane >> 4;          // 0 or 1
    const int rowbase = (blockIdx.x * WAVES + wv) * 16;
    const int koff = 2 * half;           // K sub-offset per ISA A/B layout

    // ---- Preload A (16 rows x 128), pre-scaled by 1/tau ----
    fx2 a[K_STEPS];
    const float* arow = z + (size_t)(rowbase + n) * DIM + koff;
    #pragma unroll
    for (int kk = 0; kk < K_STEPS; ++kk) {
        fx2 v = *(const fx2*)(arow + kk * 4);
        a[kk].x = v.x * INV_TAU;
        a[kk].y = v.y * INV_TAU;
    }

    // Online LSE state per (lane, accumulator-slot): logits are 10*sim
    float m[8], s[8];
    #pragma unroll
    for (int v = 0; v < 8; ++v) { m[v] = SENT; s[v] = 0.f; }

    const int npanels = B / PC;
    const float4* zv4 = (const float4*)z;   // DIM/4 = 32 float4 per row
    const int selfbase = rowbase + 8 * half;

    for (int p = 0; p < npanels; ++p) {
        const int jbase = p * PC;

        // ---- Cooperative stage: 64 cols x 128 floats -> LDS (padded) ----
        #pragma unroll 4
        for (int r = 0; r < (PC * (DIM / 4)) / BLOCK; ++r) {
            int f4 = r * BLOCK + tid;       // 0 .. 2047
            int c  = f4 >> 5;
            int kq = f4 & 31;
            float4 val = zv4[(size_t)(jbase + c) * (DIM / 4) + kq];
            *(float4*)&panel[c * PSTRIDE + kq * 4] = val;
        }
        __syncthreads();

        if (p + 1 < npanels)
            __builtin_prefetch(z + (size_t)(jbase + PC) * DIM + tid * 16, 0, 3);

        #pragma unroll
        for (int t = 0; t < PC / 16; ++t) {
            const float* bcol = &panel[(t * 16 + n) * PSTRIDE + koff];
            fx8 c8 = {0.f, 0.f, 0.f, 0.f, 0.f, 0.f, 0.f, 0.f};
            #pragma unroll
            for (int kk = 0; kk < K_STEPS; ++kk) {
                fx2 b = *(const fx2*)(bcol + kk * 4);
                c8 = __builtin_amdgcn_wmma_f32_16x16x4_f32(
                         false, a[kk], false, b, (short)0, c8, false, false);
            }

            // Branchless diagonal exclusion: element v is "self" iff v == dself.
            const int dself = (jbase + t * 16 + n) - selfbase;
            #pragma unroll
            for (int v = 0; v < 8; ++v) {
                float x  = (v == dself) ? SENT : c8[v];
                float e  = __expf(-fabsf(x - m[v]));
                bool  gt = x > m[v];
                s[v] = fmaf(s[v], gt ? e : 1.0f, gt ? 1.0f : e);
                m[v] = fmaxf(m[v], x);
            }
        }
        __syncthreads();
    }

    // ---- Positive branch via class sums (distributed dot: lane n owns dims 8n..8n+7)
    float ps[8], qs[8], np[8];
    #pragma unroll
    for (int v = 0; v < 8; ++v) {
        const int row = selfbase + v;
        const int lab = labels[row];
        const float* zr = z    + (size_t)row * DIM + n * 8;
        const float* cs = csum + (size_t)lab * DIM + n * 8;
        float pacc = 0.f, qacc = 0.f;
        #pragma unroll
        for (int d = 0; d < 8; ++d) {
            float zv = zr[d];
            pacc = fmaf(zv, cs[d], pacc);
            qacc = fmaf(zv, zv, qacc);
        }
        ps[v] = pacc; qs[v] = qacc;
        np[v] = ccnt[lab] - 1.0f;     // full value, no reduction needed
    }

    // ---- Merge 16 column-residues / dot partials per half-wave ----
    #pragma unroll
    for (int v = 0; v < 8; ++v) {
        #pragma unroll
        for (int off = 1; off < 16; off <<= 1) {
            float om = __shfl_xor(m[v],  off, 16);
            float os = __shfl_xor(s[v],  off, 16);
            float op = __shfl_xor(ps[v], off, 16);
            float oq = __shfl_xor(qs[v], off, 16);
            float nm = fmaxf(m[v], om);
            s[v] = s[v] * __expf(m[v] - nm) + os * __expf(om - nm);
            m[v] = nm; ps[v] += op; qs[v] += oq;
        }
    }

    // Lanes 0 and 16 hold full-row results for rows (rowbase + 8*half + v)
    if (n == 0) {
        float lsum = 0.f, lcnt = 0.f;
        #pragma unroll
        for (int v = 0; v < 8; ++v) {
            bool  valid = np[v] > 0.5f;
            float lse   = m[v] + __logf(s[v]);
            float pmean = (ps[v] - qs[v]) * INV_TAU / fmaxf(np[v], 1.0f);
            lsum += valid ? (lse - pmean) : 0.f;
            lcnt += valid ? 1.f : 0.f;
        }
        atomicAdd(&acc[0], lsum);
        atomicAdd(&acc[1], lcnt);
    }
}

__global__ void supcon_finalize(const float* __restrict__ acc, float* __restrict__ out)
{
    out[0] = acc[0] / fmaxf(acc[1], 1.0f);
}

extern "C" void kernel_launch(void* const* d_in, const int* in_sizes, int n_in,
                              void* d_out, int out_size, void* d_ws, size_t ws_size,
                              hipStream_t stream)
{
    const float* z      = (const float*)d_in[0];
    const int*   labels = (const int*)d_in[1];
    float*       out    = (float*)d_out;
    const int B = in_sizes[1];           // 8192

    // Workspace layout: [0..1] loss acc, [64..163] class counts, [256..13055] class sums
    float* acc  = (float*)d_ws;
    float* ccnt = acc + 64;
    float* csum = acc + 256;
    hipMemsetAsync(d_ws, 0, (256 + NC * DIM) * sizeof(float), stream);

    const int sumWGs = 32;
    class_sums<<<sumWGs, DIM, 0, stream>>>(z, labels, csum, ccnt, B / sumWGs);
    supcon_main<<<B / ROWS_PER_WG, BLOCK, 0, stream>>>(z, labels, csum, ccnt, acc, B);
    supcon_finalize<<<1, 1, 0, stream>>>(acc, out);
}